// HeteroGCNLayer_83081847374392
// MI455X (gfx1250) — compile-verified
//
#include <hip/hip_runtime.h>
#include <hip/hip_bf16.h>

// ---------------------------------------------------------------------------
// HeteroGCN layer for MI455X (gfx1250), wave32.
//   h_item = mean_agg(feat_user @ W_ui + b_ui, ui edges)          -> out[1]
//   h_user = mean_agg(feat_item @ W_iu + b_iu, iu edges)
//          + mean_agg(feat_user @ W_uu + b_uu, uu edges)          -> out[0]
// GEMMs use V_WMMA_F32_16X16X4_F32 (fp32 in/out, matches reference precision).
// ---------------------------------------------------------------------------

typedef __attribute__((ext_vector_type(2))) float v2f;
typedef __attribute__((ext_vector_type(8))) float v8f;

#define DINK 128   // K (DIN)
#define DOUTN 128  // N (DOUT)

// One block = one 16-row M tile. 8 waves, wave w owns N-tile [16w, 16w+16).
// K loop: 32 x wmma_f32_16x16x4_f32.
__global__ void __launch_bounds__(256)
hgcn_gemm_wmma(const float* __restrict__ X,    // [M, 128] row-major
               const float* __restrict__ W,    // [128, 128] row-major
               const float* __restrict__ bias, // [128]
               float* __restrict__ Y,          // [M, 128]
               int M)
{
    const int tileM = blockIdx.x;             // M / 16 tiles, M % 16 == 0
    const int wave  = threadIdx.x >> 5;       // 0..7 -> N tile
    const int lane  = threadIdx.x & 31;
    const int l16   = lane & 15;
    const int khalf = (lane >> 4) << 1;       // 0 for lanes 0-15, 2 for 16-31

    const int row = tileM * 16 + l16;         // A-matrix row for this lane
    const int col = wave * 16 + l16;          // B/N column for this lane

    const float* __restrict__ Arow = X + (size_t)row * DINK + khalf;
    const float* __restrict__ Wcol = W + col + (size_t)khalf * DOUTN;

    v8f acc = {};
#pragma unroll
    for (int kk = 0; kk < DINK / 4; ++kk) {
        const int k = kk * 4;
        // A 16x4 fp32: lanes 0-15 -> (K=k+0, K=k+1); lanes 16-31 -> (K=k+2, K=k+3)
        v2f a = *(const v2f*)(Arow + k);
        // B 4x16 fp32 (symmetric layout): N across lanes, K split by lane half
        v2f b;
        b.x = Wcol[(size_t)k * DOUTN];
        b.y = Wcol[(size_t)(k + 1) * DOUTN];
        acc = __builtin_amdgcn_wmma_f32_16x16x4_f32(
            /*neg_a=*/false, a, /*neg_b=*/false, b,
            /*c_mod=*/(short)0, acc, /*reuse_a=*/false, /*reuse_b=*/false);
    }

    const float bv = bias[col];
    // C/D layout: VGPR r -> M=r (lanes 0-15) or M=8+r (lanes 16-31); N = col
    const int mbase = tileM * 16 + ((lane >> 4) << 3);
#pragma unroll
    for (int r = 0; r < 8; ++r) {
        Y[(size_t)(mbase + r) * DOUTN + col] = acc[r] + bv;
    }
}

// One wave per edge: gather 128 floats of Wh[src], atomic-add into sum[dst].
__global__ void __launch_bounds__(256)
hgcn_scatter(const float* __restrict__ Wh,   // [Nsrc, 128]
             const int* __restrict__ src,
             const int* __restrict__ dst,
             float* __restrict__ sum,        // [Ndst, 128]
             float* __restrict__ deg,        // [Ndst]
             int E)
{
    const int gid  = blockIdx.x * blockDim.x + threadIdx.x;
    const int edge = gid >> 5;
    const int lane = gid & 31;
    if (edge >= E) return;

    const int s = src[edge];
    const int d = dst[edge];

    const float4 v = *(const float4*)(Wh + (size_t)s * DOUTN + lane * 4);
    float* p = sum + (size_t)d * DOUTN + lane * 4;
    atomicAdd(p + 0, v.x);
    atomicAdd(p + 1, v.y);
    atomicAdd(p + 2, v.z);
    atomicAdd(p + 3, v.w);
    if (lane == 0) atomicAdd(deg + d, 1.0f);
}

// mean = deg > 0 ? sum / deg : 0 ; store (add==0) or accumulate (add==1)
__global__ void __launch_bounds__(256)
hgcn_finalize(const float4* __restrict__ sum,
              const float* __restrict__ deg,
              float4* __restrict__ out,
              int N, int add)
{
    const int gid = blockIdx.x * blockDim.x + threadIdx.x;  // over N*32 float4s
    if (gid >= N * (DOUTN / 4)) return;
    const int node = gid >> 5;  // gid / 32
    const float d = deg[node];
    const float inv = (d > 0.0f) ? (1.0f / d) : 0.0f;
    float4 sv = sum[gid];
    float4 m = make_float4(sv.x * inv, sv.y * inv, sv.z * inv, sv.w * inv);
    if (add) {
        float4 o = out[gid];
        m.x += o.x; m.y += o.y; m.z += o.z; m.w += o.w;
    }
    out[gid] = m;
}

extern "C" void kernel_launch(void* const* d_in, const int* in_sizes, int n_in,
                              void* d_out, int out_size, void* d_ws, size_t ws_size,
                              hipStream_t stream) {
    const float* feat_user = (const float*)d_in[0];
    const float* feat_item = (const float*)d_in[1];
    const float* W_ui = (const float*)d_in[2];
    const float* b_ui = (const float*)d_in[3];
    const float* W_iu = (const float*)d_in[4];
    const float* b_iu = (const float*)d_in[5];
    const float* W_uu = (const float*)d_in[6];
    const float* b_uu = (const float*)d_in[7];
    const int* src_ui = (const int*)d_in[8];
    const int* dst_ui = (const int*)d_in[9];
    const int* src_iu = (const int*)d_in[10];
    const int* dst_iu = (const int*)d_in[11];
    const int* src_uu = (const int*)d_in[12];
    const int* dst_uu = (const int*)d_in[13];

    const int NU = in_sizes[0] / DINK;   // 100000
    const int NI = in_sizes[1] / DINK;   // 100000
    const int E  = in_sizes[8];          // 600000
    const int Nmax = (NU > NI) ? NU : NI;

    float* out = (float*)d_out;
    float* out_user = out;                       // out[0] = h_user
    float* out_item = out + (size_t)out_size / 2; // out[1] = h_item

    // Workspace: Wh [Nmax,128] | sum [Nmax,128] | deg [Nmax]  (sum+deg contiguous)
    float* Wh  = (float*)d_ws;
    float* sum = Wh + (size_t)Nmax * DOUTN;
    float* deg = sum + (size_t)Nmax * DOUTN;
    const size_t clear_bytes = ((size_t)Nmax * DOUTN + (size_t)Nmax) * sizeof(float);

    const dim3 blk(256);
    const dim3 grid_scat((unsigned)((E * 32 + 255) / 256));

    struct Pass {
        const float* X; const float* W; const float* b; int M;
        const int* src; const int* dst; int Ndst; float* out; int add;
    };
    const Pass passes[3] = {
        { feat_user, W_ui, b_ui, NU, src_ui, dst_ui, NI, out_item, 0 },
        { feat_item, W_iu, b_iu, NI, src_iu, dst_iu, NU, out_user, 0 },
        { feat_user, W_uu, b_uu, NU, src_uu, dst_uu, NU, out_user, 1 },
    };

    for (int p = 0; p < 3; ++p) {
        const Pass& ps = passes[p];
        // 1) projection: Wh = X @ W + b  (WMMA f32)
        hgcn_gemm_wmma<<<dim3((unsigned)(ps.M / 16)), blk, 0, stream>>>(
            ps.X, ps.W, ps.b, Wh, ps.M);
        // 2) clear accumulators
        hipMemsetAsync(sum, 0, clear_bytes, stream);
        // 3) edge scatter-add
        hgcn_scatter<<<grid_scat, blk, 0, stream>>>(Wh, ps.src, ps.dst, sum, deg, E);
        // 4) mean + write/accumulate into output
        const int nfin = ps.Ndst * (DOUTN / 4);
        hgcn_finalize<<<dim3((unsigned)((nfin + 255) / 256)), blk, 0, stream>>>(
            (const float4*)sum, deg, (float4*)ps.out, ps.Ndst, ps.add);
    }
}